// Equi_gcn2_21603685499524
// MI455X (gfx1250) — compile-verified
//
#include <hip/hip_runtime.h>

// ---------------------------------------------------------------------------
// Equi-GCN forward for MI455X (gfx1250, wave32).
// f32 end-to-end; GEMMs via v_wmma_f32_16x16x4_f32 (exact f32 matrix core).
// ---------------------------------------------------------------------------

typedef __attribute__((ext_vector_type(2))) float v2f;
typedef __attribute__((ext_vector_type(8))) float v8f;

#if defined(__has_builtin)
#if __has_builtin(__builtin_amdgcn_wmma_f32_16x16x4_f32)
#define HAVE_WMMA_F32X4 1
#endif
#endif

#define BSZ 2

// ------------------------- normalize direction kernels ---------------------
// dirs layout (A,3,S,O); normalize over the coordinate axis (axis=1).
__global__ void k_norm_dirs(const float* __restrict__ d, float* __restrict__ nd,
                            int SO, int total) {
  int t = blockIdx.x * blockDim.x + threadIdx.x;
  if (t >= total) return;
  int a = t / SO, so = t - a * SO;
  const float* p = d + (size_t)a * 3 * SO + so;
  float x = p[0], y = p[SO], z = p[2 * SO];
  float inv = 1.0f / (sqrtf(x * x + y * y + z * z) + 1e-8f);
  float* q = nd + (size_t)a * 3 * SO + so;
  q[0] = x * inv; q[SO] = y * inv; q[2 * SO] = z * inv;
}

// ------------------------------- fused KNN ---------------------------------
// One thread per query. Streams candidate tiles through LDS, keeps top-(k+1)
// smallest of (|m|^2 - 2 q.m)  (same ordering as squared distance), then
// drops the global min (self) and emits the remaining k indices (set
// semantics; all consumers are order-invariant).
#define KNN_TS 128
#define KMAXP1 17
__global__ void k_knn(const float* __restrict__ v, int npts, int qstride,
                      int nq, int ksel, int* __restrict__ outIdx) {
  __shared__ float sx[KNN_TS], sy[KNN_TS], sz[KNN_TS], sd[KNN_TS];
  int b = blockIdx.y;
  int q = blockIdx.x * blockDim.x + threadIdx.x;
  const float* vb = v + (size_t)b * npts * 3;
  bool active = q < nq;
  int qi = q * qstride;
  float qx = 0.f, qy = 0.f, qz = 0.f;
  if (active) { qx = vb[qi * 3]; qy = vb[qi * 3 + 1]; qz = vb[qi * 3 + 2]; }
  float best[KMAXP1]; int bidx[KMAXP1];
  int kp1 = ksel + 1;
  for (int i = 0; i < KMAXP1; i++) { best[i] = 3.4e38f; bidx[i] = 0; }
  float worst = 3.4e38f; int wpos = 0;
  for (int base = 0; base < npts; base += KNN_TS) {
    int m = base + threadIdx.x;           // npts is a multiple of KNN_TS
    float x = vb[m * 3], y = vb[m * 3 + 1], z = vb[m * 3 + 2];
    sx[threadIdx.x] = x; sy[threadIdx.x] = y; sz[threadIdx.x] = z;
    sd[threadIdx.x] = x * x + y * y + z * z;
    __syncthreads();
    if (active) {
      for (int j = 0; j < KNN_TS; j++) {
        float dd = sd[j] - 2.0f * (qx * sx[j] + qy * sy[j] + qz * sz[j]);
        if (dd < worst) {
          best[wpos] = dd; bidx[wpos] = base + j;
          worst = best[0]; wpos = 0;
          for (int i = 1; i < kp1; i++)
            if (best[i] > worst) { worst = best[i]; wpos = i; }
        }
      }
    }
    __syncthreads();
  }
  if (active) {
    int mpos = 0; float mval = best[0];
    for (int i = 1; i < kp1; i++) if (best[i] < mval) { mval = best[i]; mpos = i; }
    int* ob = outIdx + ((size_t)b * nq + q) * ksel;
    int w = 0;
    for (int i = 0; i < kp1; i++) if (i != mpos) ob[w++] = bidx[i];
  }
}

// ----------------------------- conv_surface --------------------------------
// fm0[b,n,a,o] = relu( sum_s max_k relu(dn[k] . nd[a,:,s,o]) ), O=32.
// Also emits apool(fm0) (max over a). Block = 128 threads = (a=4, o=32).
__global__ void k_conv_surface(const float* __restrict__ v,
                               const int* __restrict__ idx,
                               const float* __restrict__ nd,
                               float* __restrict__ fm0,
                               float* __restrict__ out0, int n) {
  __shared__ float dnx[16], dny[16], dnz[16];
  __shared__ float red[128];
  int b = blockIdx.y, i = blockIdx.x, t = threadIdx.x;
  const float* vb = v + (size_t)b * n * 3;
  if (t < 16) {
    int m = idx[((size_t)b * n + i) * 16 + t];
    float dx = vb[m * 3] - vb[i * 3];
    float dy = vb[m * 3 + 1] - vb[i * 3 + 1];
    float dz = vb[m * 3 + 2] - vb[i * 3 + 2];
    float inv = 1.0f / (sqrtf(dx * dx + dy * dy + dz * dz) + 1e-8f);
    dnx[t] = dx * inv; dny[t] = dy * inv; dnz[t] = dz * inv;
  }
  __syncthreads();
  int a = t >> 5, o = t & 31;
  float acc = 0.0f;
  for (int s = 0; s < 3; s++) {
    float n0 = nd[((a * 3 + 0) * 3 + s) * 32 + o];
    float n1 = nd[((a * 3 + 1) * 3 + s) * 32 + o];
    float n2 = nd[((a * 3 + 2) * 3 + s) * 32 + o];
    float mx = 0.0f;   // max over k of relu(.) >= 0
#pragma unroll
    for (int k = 0; k < 16; k++)
      mx = fmaxf(mx, dnx[k] * n0 + dny[k] * n1 + dnz[k] * n2);
    acc += mx;
  }
  acc = fmaxf(acc, 0.0f);
  fm0[(((size_t)b * n + i) * 4 + a) * 32 + o] = acc;
  red[t] = acc;
  __syncthreads();
  if (a == 0) {
    float m0 = fmaxf(fmaxf(red[o], red[32 + o]), fmaxf(red[64 + o], red[96 + o]));
    out0[((size_t)b * n + i) * 32 + o] = m0;
  }
}

// ------------------------------ WMMA GEMM ----------------------------------
// C[R,N] = A[R,K] @ B[K,N] + bias[N].  One wave per 16x16 tile,
// v_wmma_f32_16x16x4_f32, K-loop step 4. All dims multiples of 16 / K%4==0.
__global__ void k_gemm_wmma(const float* __restrict__ A,
                            const float* __restrict__ B,
                            const float* __restrict__ bias,
                            float* __restrict__ C, int R, int K, int N) {
  int wave = (blockIdx.x * blockDim.x + threadIdx.x) >> 5;
  int lane = threadIdx.x & 31;
  int tilesN = N >> 4;
  int tilesTotal = (R >> 4) * tilesN;
  if (wave >= tilesTotal) return;          // whole-wave uniform exit
  int tr = wave / tilesN, tc = wave - tr * tilesN;
  int rl = tr * 16 + (lane & 15);
  int c = tc * 16 + (lane & 15);
  int hi = lane >> 4;                      // K-half select (0: K0/K1, 1: K2/K3)
  v8f acc = (v8f){0.f, 0.f, 0.f, 0.f, 0.f, 0.f, 0.f, 0.f};
#if HAVE_WMMA_F32X4
  for (int kk = 0; kk < K; kk += 4) {
    const float* ap = A + (size_t)rl * K + kk + hi * 2;
    v2f av; av.x = ap[0]; av.y = ap[1];
    const float* bp = B + (size_t)(kk + hi * 2) * N + c;
    v2f bv; bv.x = bp[0]; bv.y = bp[N];
    acc = __builtin_amdgcn_wmma_f32_16x16x4_f32(false, av, false, bv,
                                                (short)0, acc, false, false);
  }
#else
  // scalar fallback with identical C/D lane mapping
  for (int kk = 0; kk < K; kk++) {
    float bvv = B[(size_t)kk * N + c];
#pragma unroll
    for (int j = 0; j < 8; j++) {
      int row = tr * 16 + j + 8 * hi;
      acc[j] += A[(size_t)row * K + kk] * bvv;
    }
  }
#endif
  float bb = bias[c];
#pragma unroll
  for (int j = 0; j < 8; j++) {
    int row = tr * 16 + j + 8 * hi;
    C[(size_t)row * N + c] = acc[j] + bb;
  }
}

// --------------------------- conv_layer activation -------------------------
// f layout (b,n,A,4O): center col o, support col O+s*O+o.
// out[b,n,a,o] = center + sum_s max_k relu(dn.nd) * sn.  Block per point.
__global__ void k_conv_act(const float* __restrict__ f,
                           const int* __restrict__ idx,
                           const float* __restrict__ v,
                           const float* __restrict__ nd,
                           float* __restrict__ out, int n, int O) {
  __shared__ float dnx[16], dny[16], dnz[16];
  __shared__ int nb[16];
  int b = blockIdx.y, i = blockIdx.x, t = threadIdx.x;
  const float* vb = v + (size_t)b * n * 3;
  size_t bn = (size_t)b * n;
  if (t < 16) {
    int m = idx[(bn + i) * 16 + t];
    nb[t] = m;
    float dx = vb[m * 3] - vb[i * 3];
    float dy = vb[m * 3 + 1] - vb[i * 3 + 1];
    float dz = vb[m * 3 + 2] - vb[i * 3 + 2];
    float inv = 1.0f / (sqrtf(dx * dx + dy * dy + dz * dz) + 1e-8f);
    dnx[t] = dx * inv; dny[t] = dy * inv; dnz[t] = dz * inv;
  }
  __syncthreads();
  int F4 = 4 * O;
  for (int pair = t; pair < 4 * O; pair += 256) {
    int a = pair / O, o = pair - a * O;
    float acc = f[((bn + i) * 4 + a) * (size_t)F4 + o];
    for (int s = 0; s < 3; s++) {
      float n0 = nd[((a * 3 + 0) * 3 + s) * O + o];
      float n1 = nd[((a * 3 + 1) * 3 + s) * O + o];
      float n2 = nd[((a * 3 + 2) * 3 + s) * O + o];
      int scol = O + s * O + o;
      float mx = -3.4e38f;
#pragma unroll
      for (int k = 0; k < 16; k++) {
        float th = fmaxf(dnx[k] * n0 + dny[k] * n1 + dnz[k] * n2, 0.0f);
        float sn = f[((bn + nb[k]) * 4 + a) * (size_t)F4 + scol];
        mx = fmaxf(mx, th * sn);
      }
      acc += mx;
    }
    out[((bn + i) * 4 + a) * (size_t)O + o] = acc;
  }
}

// ------------------------------- batchnorm ---------------------------------
__global__ void k_bn_stats(const float* __restrict__ x, float* __restrict__ mean,
                           float* __restrict__ var, int R, int O) {
  __shared__ float ss[256], sq[256];
  int c = blockIdx.x, t = threadIdx.x;
  float s = 0.f, q = 0.f;
  for (int i = t; i < R; i += 256) {
    float vv = x[(size_t)i * O + c]; s += vv; q += vv * vv;
  }
  ss[t] = s; sq[t] = q; __syncthreads();
  for (int off = 128; off > 0; off >>= 1) {
    if (t < off) { ss[t] += ss[t + off]; sq[t] += sq[t + off]; }
    __syncthreads();
  }
  if (t == 0) {
    float m = ss[0] / (float)R;
    mean[c] = m; var[c] = sq[0] / (float)R - m * m;
  }
}

__global__ void k_bn_apply_relu(float* x, const float* __restrict__ mean,
                                const float* __restrict__ var,
                                const float* __restrict__ g,
                                const float* __restrict__ be, int total, int O) {
  int e = blockIdx.x * blockDim.x + threadIdx.x;
  if (e >= total) return;
  int c = e % O;
  float y = g[c] * (x[e] - mean[c]) * rsqrtf(var[c] + 1e-5f) + be[c];
  x[e] = fmaxf(y, 0.0f);
}

// --------------------------------- apool -----------------------------------
__global__ void k_apool(const float* __restrict__ fm, float* __restrict__ out,
                        int rows, int O) {
  int e = blockIdx.x * blockDim.x + threadIdx.x;
  if (e >= rows * O) return;
  int r = e / O, o = e - r * O;
  const float* p = fm + (size_t)r * 4 * O + o;
  out[e] = fmaxf(fmaxf(p[0], p[O]), fmaxf(p[2 * O], p[3 * O]));
}

// ------------------------------ pool helpers -------------------------------
__global__ void k_pool_gather(const float* __restrict__ fm,
                              const int* __restrict__ idxp,
                              float* __restrict__ fp, int nq, int O, int npts) {
  int e = blockIdx.x * blockDim.x + threadIdx.x;
  int total = BSZ * nq * 4 * O;
  if (e >= total) return;
  int o = e % O;
  int a = (e / O) & 3;
  int j = (e / (4 * O)) % nq;
  int b = e / (4 * O * nq);
  const int* ip = idxp + ((size_t)b * nq + j) * 4;
  float m = -3.4e38f;
#pragma unroll
  for (int k = 0; k < 4; k++)
    m = fmaxf(m, fm[(((size_t)b * npts + ip[k]) * 4 + a) * O + o]);
  fp[e] = m;
}

__global__ void k_subsample_v(const float* __restrict__ v, float* __restrict__ vout,
                              int nq, int rate, int npts) {
  int e = blockIdx.x * blockDim.x + threadIdx.x;
  if (e >= BSZ * nq * 3) return;
  int d = e % 3;
  int j = (e / 3) % nq;
  int b = e / (3 * nq);
  vout[e] = v[((size_t)b * npts + j * rate) * 3 + d];
}

// ============================ host orchestration ============================
enum { P_D0 = 0, P_W1, P_B1, P_D1, P_W2, P_B2, P_D2, P_W3, P_B3, P_D3,
       P_W4, P_B4, P_D4, P_G1, P_BE1, P_G2, P_BE2, P_G3, P_BE3 };

static inline unsigned cdiv(unsigned a, unsigned b) { return (a + b - 1) / b; }

extern "C" void kernel_launch(void* const* d_in, const int* in_sizes, int n_in,
                              void* d_out, int out_size, void* d_ws, size_t ws_size,
                              hipStream_t stream) {
  // ---- map inputs (robust to flatten ordering) ----
  const float* P[19] = {nullptr};
  const float* verts = nullptr;
  int vi = -1;
  for (int i = 0; i < n_in; i++)
    if (in_sizes[i] == BSZ * 4096 * 3) { vi = i; break; }
  if (vi < 0) vi = 0;
  verts = (const float*)d_in[vi];
  int slots[32]; int w = 0;
  for (int i = 0; i < n_in && w < 19; i++) if (i != vi) slots[w++] = i;
  // canonical (insertion) order: d0,W1,b1,d1,W2,b2,d2,W3,b3,d3,W4,b4,d4,g1,be1,g2,be2,g3,be3
  // sorted-key order pos for each canonical index:
  static const int sorted_pos[19] = {11, 0, 4, 12, 1, 5, 13, 2, 6, 14,
                                     3, 7, 15, 16, 8, 17, 9, 18, 10};
  bool sorted = (w > 0) && (in_sizes[slots[0]] == 8192); // W1 first => sorted
  for (int ci = 0; ci < 19; ci++)
    P[ci] = (const float*)d_in[slots[sorted ? sorted_pos[ci] : ci]];

  // ---- workspace layout (float units) ----
  float* wsf = (float*)d_ws;
  const size_t OFF_ND0 = 0, OFF_ND1 = 1152, OFF_ND2 = 3456, OFF_ND3 = 8064,
               OFF_ND4 = 17280, OFF_MEAN = 35712, OFF_VAR = 35968,
               OFF_IDX1 = 36224, OFF_IDXP1 = 167296, OFF_IDX2 = 175488,
               OFF_IDXP2 = 208256, OFF_IDX3 = 210304, OFF_FM0 = 214400,
               OFF_F13 = 1262976, OFF_FM1 = 9651584, OFF_FP1 = 11748736,
               OFF_F24 = 12273024, OFF_FM2 = 16467328, OFF_FM3 = 17515904,
               OFF_FP2 = 19613056, OFF_FM4 = 20137344, WS_END = 21185920;
  if (ws_size < WS_END * sizeof(float)) return;

  float* nd0 = wsf + OFF_ND0; float* nd1 = wsf + OFF_ND1;
  float* nd2 = wsf + OFF_ND2; float* nd3 = wsf + OFF_ND3;
  float* nd4 = wsf + OFF_ND4;
  float* bmean = wsf + OFF_MEAN; float* bvar = wsf + OFF_VAR;
  int* idx1 = (int*)(wsf + OFF_IDX1);
  int* idxp1 = (int*)(wsf + OFF_IDXP1);
  int* idx2 = (int*)(wsf + OFF_IDX2);
  int* idxp2 = (int*)(wsf + OFF_IDXP2);
  int* idx3 = (int*)(wsf + OFF_IDX3);
  float* fm0 = wsf + OFF_FM0;
  float* f13 = wsf + OFF_F13;   // arena for f1 (b,4096,4,256) and f3 (b,1024,4,1024)
  float* fm1 = wsf + OFF_FM1;
  float* fp1 = wsf + OFF_FP1;
  float* f24 = wsf + OFF_F24;   // arena for f2 (b,1024,4,512) and f4 (b,256,4,2048)
  float* fm2 = wsf + OFF_FM2;
  float* fm3 = wsf + OFF_FM3;
  float* fp2 = wsf + OFF_FP2;
  float* fm4 = wsf + OFF_FM4;

  float* out = (float*)d_out;
  float* out0 = out;                 // (2,4096,32)
  float* out1 = out + 262144;       // (2,4096,64)
  float* out2 = out + 786432;       // (2,1024,128)
  float* out3 = out + 1048576;      // (2,1024,256)
  float* out4 = out + 1572864;      // (2,256,512)
  float* v1o = out + 1835008;       // (2,1024,3)
  float* v2o = out + 1841152;       // (2,256,3)

  // ---- normalize direction kernels ----
  struct { const float* src; float* dst; int O; } nds[5] = {
      {P[P_D0], nd0, 32}, {P[P_D1], nd1, 64}, {P[P_D2], nd2, 128},
      {P[P_D3], nd3, 256}, {P[P_D4], nd4, 512}};
  for (int i = 0; i < 5; i++) {
    int total = 4 * 3 * nds[i].O;
    k_norm_dirs<<<cdiv(total, 256), 256, 0, stream>>>(nds[i].src, nds[i].dst,
                                                      3 * nds[i].O, total);
  }

  // ---- stage 0: KNN + conv_surface ----
  k_knn<<<dim3(4096 / 128, BSZ), 128, 0, stream>>>(verts, 4096, 1, 4096, 16, idx1);
  k_conv_surface<<<dim3(4096, BSZ), 128, 0, stream>>>(verts, idx1, nd0, fm0, out0, 4096);

  // ---- layer 1 ----
  k_gemm_wmma<<<cdiv((32768 / 16) * (256 / 16), 8), 256, 0, stream>>>(
      fm0, P[P_W1], P[P_B1], f13, 32768, 32, 256);
  k_conv_act<<<dim3(4096, BSZ), 256, 0, stream>>>(f13, idx1, verts, nd1, fm1, 4096, 64);
  k_bn_stats<<<64, 256, 0, stream>>>(fm1, bmean, bvar, 32768, 64);
  k_bn_apply_relu<<<cdiv(32768 * 64, 256), 256, 0, stream>>>(fm1, bmean, bvar,
      P[P_G1], P[P_BE1], 32768 * 64, 64);
  k_apool<<<cdiv(8192 * 64, 256), 256, 0, stream>>>(fm1, out1, 8192, 64);

  // ---- pool 1 ----
  k_knn<<<dim3(1024 / 128, BSZ), 128, 0, stream>>>(verts, 4096, 4, 1024, 4, idxp1);
  k_pool_gather<<<cdiv(BSZ * 1024 * 4 * 64, 256), 256, 0, stream>>>(
      fm1, idxp1, fp1, 1024, 64, 4096);
  k_subsample_v<<<cdiv(BSZ * 1024 * 3, 256), 256, 0, stream>>>(verts, v1o, 1024, 4, 4096);

  // ---- layer 2 ----
  k_knn<<<dim3(1024 / 128, BSZ), 128, 0, stream>>>(v1o, 1024, 1, 1024, 16, idx2);
  k_gemm_wmma<<<cdiv((8192 / 16) * (512 / 16), 8), 256, 0, stream>>>(
      fp1, P[P_W2], P[P_B2], f24, 8192, 64, 512);
  k_conv_act<<<dim3(1024, BSZ), 256, 0, stream>>>(f24, idx2, v1o, nd2, fm2, 1024, 128);
  k_bn_stats<<<128, 256, 0, stream>>>(fm2, bmean, bvar, 8192, 128);
  k_bn_apply_relu<<<cdiv(8192 * 128, 256), 256, 0, stream>>>(fm2, bmean, bvar,
      P[P_G2], P[P_BE2], 8192 * 128, 128);
  k_apool<<<cdiv(2048 * 128, 256), 256, 0, stream>>>(fm2, out2, 2048, 128);

  // ---- layer 3 ----
  k_gemm_wmma<<<cdiv((8192 / 16) * (1024 / 16), 8), 256, 0, stream>>>(
      fm2, P[P_W3], P[P_B3], f13, 8192, 128, 1024);
  k_conv_act<<<dim3(1024, BSZ), 256, 0, stream>>>(f13, idx2, v1o, nd3, fm3, 1024, 256);
  k_bn_stats<<<256, 256, 0, stream>>>(fm3, bmean, bvar, 8192, 256);
  k_bn_apply_relu<<<cdiv(8192 * 256, 256), 256, 0, stream>>>(fm3, bmean, bvar,
      P[P_G3], P[P_BE3], 8192 * 256, 256);
  k_apool<<<cdiv(2048 * 256, 256), 256, 0, stream>>>(fm3, out3, 2048, 256);

  // ---- pool 2 ----
  k_knn<<<dim3(256 / 128, BSZ), 128, 0, stream>>>(v1o, 1024, 4, 256, 4, idxp2);
  k_pool_gather<<<cdiv(BSZ * 256 * 4 * 256, 256), 256, 0, stream>>>(
      fm3, idxp2, fp2, 256, 256, 1024);
  k_subsample_v<<<cdiv(BSZ * 256 * 3, 256), 256, 0, stream>>>(v1o, v2o, 256, 4, 1024);

  // ---- layer 4 (no BN / no relu) ----
  k_knn<<<dim3(256 / 128, BSZ), 128, 0, stream>>>(v2o, 256, 1, 256, 16, idx3);
  k_gemm_wmma<<<cdiv((2048 / 16) * (2048 / 16), 8), 256, 0, stream>>>(
      fp2, P[P_W4], P[P_B4], f24, 2048, 256, 2048);
  k_conv_act<<<dim3(256, BSZ), 256, 0, stream>>>(f24, idx3, v2o, nd4, fm4, 256, 512);
  k_apool<<<cdiv(512 * 512, 256), 256, 0, stream>>>(fm4, out4, 512, 512);
}